// FlexibleThresholdedLoss_49478023249977
// MI455X (gfx1250) — compile-verified
//
#include <hip/hip_runtime.h>
#include <math.h>

// ---------------------------------------------------------------------------
// FlexibleThresholdedLoss for MI455X (gfx1250, wave32).
// Memory-bound two-pass reduction:
//   pass1: sum|d|, sum d^2  -> mae, mse means
//   pass2: masked sums/counts with thresholds mae/mse
// ~201MB working set vs 192MB L2: pass2 largely L2-resident with default RT
// temporal hints. Wave-level reduction uses v_wmma_f32_16x16x4_f32 with an
// all-ones B matrix (D[m][n] = sum_k A[m][k] + C[m][n]) to collapse the
// 4 per-lane f32 accumulator chains of a wave on the matrix pipe.
// ---------------------------------------------------------------------------

typedef __attribute__((ext_vector_type(2))) float v2f;
typedef __attribute__((ext_vector_type(8))) float v8f;

#define NTHREADS 256
#define NBLK_MAX 2048

// Reduce 4 f32 accumulators per lane across the whole wave32.
// Two chained WMMAs (C-accumulate) put row sums of the 16x4 A tile in D.
// D layout (32-bit C/D): VGPR j, lanes 0-15 -> M=j, lanes 16-31 -> M=8+j,
// every column N identical (B == ones). Sum VGPRs 0..7 per lane, then add
// the complementary half-wave via shfl_xor(16): every lane holds the total.
__device__ __forceinline__ float wave_sum4_wmma(float a0, float a1, float a2, float a3) {
    v2f A, B;
    v8f C = {0.f, 0.f, 0.f, 0.f, 0.f, 0.f, 0.f, 0.f};
    B[0] = 1.0f; B[1] = 1.0f;          // all-ones B: layout-independent
    A[0] = a0;  A[1] = a1;
    C = __builtin_amdgcn_wmma_f32_16x16x4_f32(false, A, false, B, (short)0, C, false, false);
    A[0] = a2;  A[1] = a3;
    C = __builtin_amdgcn_wmma_f32_16x16x4_f32(false, A, false, B, (short)0, C, false, false);
    float t = ((C[0] + C[1]) + (C[2] + C[3])) + ((C[4] + C[5]) + (C[6] + C[7]));
    t += __shfl_xor(t, 16, 32);        // pair half-waves -> full wave sum
    return t;
}

// ---------------------------------------------------------------------------
// Pass 1: per-block partial sum|d| and sum d^2.
// ---------------------------------------------------------------------------
__global__ void __launch_bounds__(NTHREADS)
pass1_kernel(const float* __restrict__ a, const float* __restrict__ b,
             float* __restrict__ p_abs, float* __restrict__ p_sq, long long n) {
    const long long n4 = n >> 2;
    const float4* __restrict__ a4 = (const float4*)a;
    const float4* __restrict__ b4 = (const float4*)b;

    float sa0 = 0.f, sa1 = 0.f, sa2 = 0.f, sa3 = 0.f;
    float sq0 = 0.f, sq1 = 0.f, sq2 = 0.f, sq3 = 0.f;

    const long long stride = (long long)gridDim.x * blockDim.x;
    for (long long i = (long long)blockIdx.x * blockDim.x + threadIdx.x; i < n4; i += stride) {
        float4 x = a4[i];
        float4 y = b4[i];
        float d0 = x.x - y.x, d1 = x.y - y.y, d2 = x.z - y.z, d3 = x.w - y.w;
        sa0 += fabsf(d0); sa1 += fabsf(d1); sa2 += fabsf(d2); sa3 += fabsf(d3);
        sq0 = fmaf(d0, d0, sq0); sq1 = fmaf(d1, d1, sq1);
        sq2 = fmaf(d2, d2, sq2); sq3 = fmaf(d3, d3, sq3);
    }
    if (blockIdx.x == 0) {             // scalar tail (N%4), converges before WMMA
        for (long long i = n4 * 4 + threadIdx.x; i < n; i += blockDim.x) {
            float d = a[i] - b[i];
            sa0 += fabsf(d);
            sq0 = fmaf(d, d, sq0);
        }
    }

    float wa = wave_sum4_wmma(sa0, sa1, sa2, sa3);
    float wq = wave_sum4_wmma(sq0, sq1, sq2, sq3);

    __shared__ float lds[2][8];
    const int wave = threadIdx.x >> 5, lane = threadIdx.x & 31;
    if (lane == 0) { lds[0][wave] = wa; lds[1][wave] = wq; }
    __syncthreads();
    if (threadIdx.x == 0) {
        float ta = 0.f, tq = 0.f;
        const int nw = blockDim.x >> 5;
        for (int w = 0; w < nw; ++w) { ta += lds[0][w]; tq += lds[1][w]; }
        p_abs[blockIdx.x] = ta;
        p_sq[blockIdx.x]  = tq;
    }
}

// ---------------------------------------------------------------------------
// Reduce pass-1 partials -> scal[0]=mae, scal[1]=mse (single block).
// ---------------------------------------------------------------------------
__global__ void __launch_bounds__(NTHREADS)
means_kernel(const float* __restrict__ p_abs, const float* __restrict__ p_sq,
             int nblk, long long n, float* __restrict__ scal) {
    float sa = 0.f, sq = 0.f;
    for (int i = threadIdx.x; i < nblk; i += blockDim.x) { sa += p_abs[i]; sq += p_sq[i]; }
    float wa = wave_sum4_wmma(sa, 0.f, 0.f, 0.f);
    float wq = wave_sum4_wmma(sq, 0.f, 0.f, 0.f);

    __shared__ float lds[2][8];
    const int wave = threadIdx.x >> 5, lane = threadIdx.x & 31;
    if (lane == 0) { lds[0][wave] = wa; lds[1][wave] = wq; }
    __syncthreads();
    if (threadIdx.x == 0) {
        float ta = 0.f, tq = 0.f;
        const int nw = blockDim.x >> 5;
        for (int w = 0; w < nw; ++w) { ta += lds[0][w]; tq += lds[1][w]; }
        const float inv_n_den = (float)n;          // 25165824 is exact in f32
        scal[0] = ta / inv_n_den;                  // mae_loss
        scal[1] = tq / inv_n_den;                  // mse_loss
    }
}

// ---------------------------------------------------------------------------
// Pass 2: masked sums + counts per block (inputs now mostly L2-resident).
// ---------------------------------------------------------------------------
__global__ void __launch_bounds__(NTHREADS)
pass2_kernel(const float* __restrict__ a, const float* __restrict__ b,
             const float* __restrict__ scal,
             float* __restrict__ p_sabs, float* __restrict__ p_ssq,
             float* __restrict__ p_cabs, float* __restrict__ p_csq, long long n) {
    const float mae = scal[0];
    const float mse = scal[1];
    const long long n4 = n >> 2;
    const float4* __restrict__ a4 = (const float4*)a;
    const float4* __restrict__ b4 = (const float4*)b;

    float ma0 = 0.f, ma1 = 0.f, ma2 = 0.f, ma3 = 0.f;   // masked |d| sums
    float ms0 = 0.f, ms1 = 0.f, ms2 = 0.f, ms3 = 0.f;   // masked d^2 sums
    float ca0 = 0.f, ca1 = 0.f, ca2 = 0.f, ca3 = 0.f;   // |d| mask counts
    float cs0 = 0.f, cs1 = 0.f, cs2 = 0.f, cs3 = 0.f;   // d^2 mask counts

    const long long stride = (long long)gridDim.x * blockDim.x;
    for (long long i = (long long)blockIdx.x * blockDim.x + threadIdx.x; i < n4; i += stride) {
        float4 x = a4[i];
        float4 y = b4[i];
        float d0 = x.x - y.x, d1 = x.y - y.y, d2 = x.z - y.z, d3 = x.w - y.w;
        float A0 = fabsf(d0), A1 = fabsf(d1), A2 = fabsf(d2), A3 = fabsf(d3);
        float S0 = d0 * d0,  S1 = d1 * d1,  S2 = d2 * d2,  S3 = d3 * d3;
        if (A0 >= mae) { ma0 += A0; ca0 += 1.f; }
        if (A1 >= mae) { ma1 += A1; ca1 += 1.f; }
        if (A2 >= mae) { ma2 += A2; ca2 += 1.f; }
        if (A3 >= mae) { ma3 += A3; ca3 += 1.f; }
        if (S0 >= mse) { ms0 += S0; cs0 += 1.f; }
        if (S1 >= mse) { ms1 += S1; cs1 += 1.f; }
        if (S2 >= mse) { ms2 += S2; cs2 += 1.f; }
        if (S3 >= mse) { ms3 += S3; cs3 += 1.f; }
    }
    if (blockIdx.x == 0) {
        for (long long i = n4 * 4 + threadIdx.x; i < n; i += blockDim.x) {
            float d = a[i] - b[i];
            float A = fabsf(d), S = d * d;
            if (A >= mae) { ma0 += A; ca0 += 1.f; }
            if (S >= mse) { ms0 += S; cs0 += 1.f; }
        }
    }

    float wsa = wave_sum4_wmma(ma0, ma1, ma2, ma3);
    float wss = wave_sum4_wmma(ms0, ms1, ms2, ms3);
    float wca = wave_sum4_wmma(ca0, ca1, ca2, ca3);
    float wcs = wave_sum4_wmma(cs0, cs1, cs2, cs3);

    __shared__ float lds[4][8];
    const int wave = threadIdx.x >> 5, lane = threadIdx.x & 31;
    if (lane == 0) { lds[0][wave] = wsa; lds[1][wave] = wss; lds[2][wave] = wca; lds[3][wave] = wcs; }
    __syncthreads();
    if (threadIdx.x == 0) {
        float t0 = 0.f, t1 = 0.f, t2 = 0.f, t3 = 0.f;
        const int nw = blockDim.x >> 5;
        for (int w = 0; w < nw; ++w) { t0 += lds[0][w]; t1 += lds[1][w]; t2 += lds[2][w]; t3 += lds[3][w]; }
        p_sabs[blockIdx.x] = t0;   // per-block counts <= ~16K: exact in f32
        p_ssq[blockIdx.x]  = t1;
        p_cabs[blockIdx.x] = t2;
        p_csq[blockIdx.x]  = t3;
    }
}

// ---------------------------------------------------------------------------
// Final: reduce pass-2 partials, combine with weights (single block).
// ---------------------------------------------------------------------------
__global__ void __launch_bounds__(NTHREADS)
final_kernel(const float* __restrict__ p_sabs, const float* __restrict__ p_ssq,
             const float* __restrict__ p_cabs, const float* __restrict__ p_csq,
             const float* __restrict__ scal, int nblk, float* __restrict__ out) {
    float sa = 0.f, sq = 0.f, ca = 0.f, cq = 0.f;
    for (int i = threadIdx.x; i < nblk; i += blockDim.x) {
        sa += p_sabs[i]; sq += p_ssq[i]; ca += p_cabs[i]; cq += p_csq[i];
    }
    float wsa = wave_sum4_wmma(sa, 0.f, 0.f, 0.f);
    float wsq = wave_sum4_wmma(sq, 0.f, 0.f, 0.f);
    float wca = wave_sum4_wmma(ca, 0.f, 0.f, 0.f);
    float wcq = wave_sum4_wmma(cq, 0.f, 0.f, 0.f);

    __shared__ float lds[4][8];
    const int wave = threadIdx.x >> 5, lane = threadIdx.x & 31;
    if (lane == 0) { lds[0][wave] = wsa; lds[1][wave] = wsq; lds[2][wave] = wca; lds[3][wave] = wcq; }
    __syncthreads();
    if (threadIdx.x == 0) {
        float  tsa = 0.f, tsq = 0.f;
        double tca = 0.0, tcq = 0.0;   // total counts up to 25.2M > 2^24: sum in f64
        const int nw = blockDim.x >> 5;
        for (int w = 0; w < nw; ++w) {
            tsa += lds[0][w]; tsq += lds[1][w];
            tca += (double)lds[2][w]; tcq += (double)lds[3][w];
        }
        const float mae = scal[0];
        const float mse = scal[1];
        const float mae_thr = (tca > 0.0) ? (tsa / fmaxf((float)tca, 1.0f)) : 0.0f;
        const float mse_thr = (tcq > 0.0) ? (tsq / fmaxf((float)tcq, 1.0f)) : 0.0f;
        const float comb_thr     = 0.5f * mae_thr + 0.5f * mse_thr;   // L1_W, L2_W
        const float comb_non_thr = 0.5f * mae     + 0.5f * mse;
        out[0] = 0.5f * comb_thr + 0.5f * comb_non_thr;               // THR_W, NON_THR_W
    }
}

// ---------------------------------------------------------------------------
// Launch
// ---------------------------------------------------------------------------
extern "C" void kernel_launch(void* const* d_in, const int* in_sizes, int n_in,
                              void* d_out, int out_size, void* d_ws, size_t ws_size,
                              hipStream_t stream) {
    const float* a = (const float*)d_in[0];
    const float* b = (const float*)d_in[1];
    const long long n = (long long)in_sizes[0];

    // Workspace layout (floats): [0..1]=scalars(mae,mse), then 6 arrays of nblk.
    float* ws = (float*)d_ws;
    long long avail = (long long)(ws_size / sizeof(float));
    int nblk = (int)((avail - 2) / 6);
    if (nblk > NBLK_MAX) nblk = NBLK_MAX;
    if (nblk < 1) nblk = 1;

    float* scal = ws;
    float* p1a  = ws + 2;
    float* p1q  = p1a + nblk;
    float* p2sa = p1q + nblk;
    float* p2sq = p2sa + nblk;
    float* p2ca = p2sq + nblk;
    float* p2cq = p2ca + nblk;

    pass1_kernel<<<nblk, NTHREADS, 0, stream>>>(a, b, p1a, p1q, n);
    means_kernel<<<1, NTHREADS, 0, stream>>>(p1a, p1q, nblk, n, scal);
    pass2_kernel<<<nblk, NTHREADS, 0, stream>>>(a, b, scal, p2sa, p2sq, p2ca, p2cq, n);
    final_kernel<<<1, NTHREADS, 0, stream>>>(p2sa, p2sq, p2ca, p2cq, scal, nblk, (float*)d_out);
}